// FeatureGrid_39178691674399
// MI455X (gfx1250) — compile-verified
//
#include <hip/hip_runtime.h>

#define RES   128
#define FEAT  32          // == wave32 lane count: lane = feature channel
#define PTS_PER_WAVE 8
#define BLOCK 256         // 8 waves -> 64 points per block

__device__ __forceinline__ int clampi(int v, int lo, int hi) {
    return v < lo ? lo : (v > hi ? hi : v);
}

// one point's trilinear gather + weighted sum; all scalar math is wave-uniform
__device__ __forceinline__ void do_point(int n, int lane,
                                         const float* __restrict__ pt,
                                         const float* __restrict__ grid,
                                         float* __restrict__ out)
{
    const float px = pt[(size_t)n * 3 + 0];
    const float py = pt[(size_t)n * 3 + 1];
    const float pz = pt[(size_t)n * 3 + 2];

    // alpha = (p + 1)/2 ; idx_f = alpha * (RES-1)
    const float fx = ((px + 1.0f) * 0.5f) * 127.0f;
    const float fy = ((py + 1.0f) * 0.5f) * 127.0f;
    const float fz = ((pz + 1.0f) * 0.5f) * 127.0f;

    const int ix = (int)fx;               // trunc == astype(int32)
    const int iy = (int)fy;
    const int iz = (int)fz;

    const float ax = fx - (float)ix;      // modf fractional part
    const float ay = fy - (float)iy;
    const float az = fz - (float)iz;

    const int x0 = clampi(ix,     0, RES - 1);
    const int x1 = clampi(ix + 1, 0, RES - 1);
    const int y0 = clampi(iy,     0, RES - 1);
    const int y1 = clampi(iy + 1, 0, RES - 1);
    const int z0 = clampi(iz,     0, RES - 1);
    const int z1 = clampi(iz + 1, 0, RES - 1);

    const float bx0 = 1.0f - ax, bx1 = ax;
    const float by0 = 1.0f - ay, by1 = ay;
    const float bz0 = 1.0f - az, bz1 = az;

    const float w00 = bx0 * by0, w01 = bx0 * by1;
    const float w10 = bx1 * by0, w11 = bx1 * by1;

    const int rx0 = x0 * (RES * RES), rx1 = x1 * (RES * RES);
    const int ry0 = y0 * RES,         ry1 = y1 * RES;

    const float* __restrict__ g = grid + lane;   // lane = feature channel
    const int o000 = (rx0 + ry0 + z0) * FEAT;
    const int o001 = (rx0 + ry0 + z1) * FEAT;
    const int o010 = (rx0 + ry1 + z0) * FEAT;
    const int o011 = (rx0 + ry1 + z1) * FEAT;
    const int o100 = (rx1 + ry0 + z0) * FEAT;
    const int o101 = (rx1 + ry0 + z1) * FEAT;
    const int o110 = (rx1 + ry1 + z0) * FEAT;
    const int o111 = (rx1 + ry1 + z1) * FEAT;

    // 8 perfectly-coalesced 128B gathers (one cache line per wave each)
    const float v000 = g[o000];
    const float v001 = g[o001];
    const float v010 = g[o010];
    const float v011 = g[o011];
    const float v100 = g[o100];
    const float v101 = g[o101];
    const float v110 = g[o110];
    const float v111 = g[o111];

    float acc;
    acc = v000 * (w00 * bz0);
    acc = fmaf(v001, w00 * bz1, acc);
    acc = fmaf(v010, w01 * bz0, acc);
    acc = fmaf(v011, w01 * bz1, acc);
    acc = fmaf(v100, w10 * bz0, acc);
    acc = fmaf(v101, w10 * bz1, acc);
    acc = fmaf(v110, w11 * bz0, acc);
    acc = fmaf(v111, w11 * bz1, acc);

    // write-once 128MB output stream: keep it out of the 192MB L2 (grid wants L2)
    __builtin_nontemporal_store(acc, out + (size_t)n * FEAT + lane);
}

__global__ __launch_bounds__(BLOCK, 4)
void FeatureGrid_trilerp_kernel(const float* __restrict__ pt,
                                const float* __restrict__ grid,
                                float* __restrict__ out,
                                int n_pts)
{
    const int lane = (int)(threadIdx.x & 31u);
    // force wave id into an SGPR so per-point scalar math/loads scalarize
    const int wave = __builtin_amdgcn_readfirstlane((int)(threadIdx.x >> 5));
    const int wavesPerBlock = (int)(blockDim.x >> 5);
    const int base = ((int)blockIdx.x * wavesPerBlock + wave) * PTS_PER_WAVE;

    // streaming prefetch of the next tile of point coords (global_prefetch_b8)
    __builtin_prefetch(pt + (size_t)(base + PTS_PER_WAVE) * 3, 0, 0);

    if (base + PTS_PER_WAVE <= n_pts) {
        // fast path: no bounds checks between load clusters -> compiler can
        // batch all 64 gathers of this wave into long back-to-back runs
#pragma unroll
        for (int i = 0; i < PTS_PER_WAVE; ++i)
            do_point(base + i, lane, pt, grid, out);
    } else {
        // tail path (taken by at most one wave-tile)
#pragma unroll
        for (int i = 0; i < PTS_PER_WAVE; ++i) {
            const int n = base + i;
            if (n >= n_pts) return;   // wave-uniform branch
            do_point(n, lane, pt, grid, out);
        }
    }
}

extern "C" void kernel_launch(void* const* d_in, const int* in_sizes, int n_in,
                              void* d_out, int out_size, void* d_ws, size_t ws_size,
                              hipStream_t stream)
{
    const float* pt   = (const float*)d_in[0];   // [N_PTS, 3] f32
    const float* grid = (const float*)d_in[1];   // [128,128,128,32] f32
    float*       out  = (float*)d_out;           // [N_PTS, 32] f32

    const int n_pts = in_sizes[0] / 3;
    const int ptsPerBlock = (BLOCK / 32) * PTS_PER_WAVE;   // 64
    const int blocks = (n_pts + ptsPerBlock - 1) / ptsPerBlock;

    FeatureGrid_trilerp_kernel<<<blocks, BLOCK, 0, stream>>>(pt, grid, out, n_pts);
}